// LightGCN_SimGCL_12257836662828
// MI455X (gfx1250) — compile-verified
//
#include <hip/hip_runtime.h>
#include <hip/hip_bf16.h>

// LightGCN propagation on MI455X (gfx1250).
//
// Roofline: 0.77 GFLOP vs ~3 GB of gather/atomic traffic -> purely memory
// bound. The 38.4 MB embedding table is L2-resident (192 MB L2), so we
// optimize for L2 bandwidth + L2 f32 atomic throughput:
//   * 16 lanes/edge, float4 per lane -> 256B coalesced row gathers
//   * global_atomic_add_f32 scatter (L2-side RMW)
//   * per-block 256-edge metadata tile staged into LDS with the CDNA5
//     async-to-LDS path (ASYNCcnt), read back via LDS broadcast
//   * next-layer buffer zeroing fused into already-bandwidth-bound passes
//     (no standalone 38.4 MB zero pass; 7 launches total).

#define EMB        64
#define TILE       256      // edges staged per block
#define LANES_PE   16       // lanes per edge (16 * float4 = 64 floats)
#define BLOCK      256

#if defined(__gfx1250__) && \
    __has_builtin(__builtin_amdgcn_global_load_async_to_lds_b32) && \
    __has_builtin(__builtin_amdgcn_s_wait_asynccnt)
#define USE_ASYNC_LDS 1
#else
#define USE_ASYNC_LDS 0
#endif

// ---------------------------------------------------------------------------
// acc = x = concat(user_emb, item_emb); zero the first scatter target.
// ---------------------------------------------------------------------------
__global__ void lgcn_init_k(const float4* __restrict__ ue,
                            const float4* __restrict__ ie,
                            float4* __restrict__ x,
                            float4* __restrict__ acc,
                            float4* __restrict__ zbuf,
                            int nu4, int nt4) {
    int i = blockIdx.x * blockDim.x + threadIdx.x;
    if (i >= nt4) return;
    const float4 v = (i < nu4) ? ue[i] : ie[i - nu4];
    float4 o;
    o.x = v.x; o.y = v.y; o.z = v.z; o.w = v.w;
    x[i]    = o;
    acc[i]  = o;
    zbuf[i] = make_float4(0.f, 0.f, 0.f, 0.f);
}

// ---------------------------------------------------------------------------
// acc = (acc + xn) * s; also zero the (dead) buffer the NEXT layer scatters
// into, fused into this already-running bandwidth pass. zbuf==nullptr on the
// last layer (nothing left to zero).
// ---------------------------------------------------------------------------
__global__ void lgcn_addzero_k(float4* __restrict__ acc,
                               const float4* __restrict__ xn,
                               float4* __restrict__ zbuf,
                               float s, int n4) {
    int i = blockIdx.x * blockDim.x + threadIdx.x;
    if (i >= n4) return;
    float4 a = acc[i];
    const float4 b = xn[i];
    a.x = (a.x + b.x) * s;
    a.y = (a.y + b.y) * s;
    a.z = (a.z + b.z) * s;
    a.w = (a.w + b.w) * s;
    acc[i] = a;
    if (zbuf) zbuf[i] = make_float4(0.f, 0.f, 0.f, 0.f);
}

// ---------------------------------------------------------------------------
// Scatter: for each edge e: xn[row[e]] += x[col[e]] * w[e]
// Block handles a TILE of 256 edges. Edge metadata is staged into LDS with
// gfx1250 async loads, then 8 waves sweep the tile: each group of 16 lanes
// owns one edge (float4 per lane -> fully coalesced 256B row transfers).
// ---------------------------------------------------------------------------
__global__ void lgcn_scatter_k(const int*   __restrict__ row,
                               const int*   __restrict__ col,
                               const float* __restrict__ w,
                               const float* __restrict__ x,
                               float*       __restrict__ xn,
                               int E) {
    __shared__ int   rowS[TILE];
    __shared__ int   colS[TILE];
    __shared__ float wS[TILE];

    const int t    = threadIdx.x;
    const int base = blockIdx.x * TILE;
    const int g    = base + t;

#if USE_ASYNC_LDS
    if (g < E) {
        __builtin_amdgcn_global_load_async_to_lds_b32(
            (int*)(row + g), (int*)(rowS + t), 0, 0);
        __builtin_amdgcn_global_load_async_to_lds_b32(
            (int*)(col + g), (int*)(colS + t), 0, 0);
        __builtin_amdgcn_global_load_async_to_lds_b32(
            (int*)(w + g),   (int*)(wS + t),   0, 0);
    }
    __builtin_amdgcn_s_wait_asynccnt(0);
#else
    if (g < E) {
        rowS[t] = row[g];
        colS[t] = col[g];
        wS[t]   = w[g];
    }
#endif
    __syncthreads();

    const int sub   = t & (LANES_PE - 1);   // which float4 of the row
    const int eLane = t >> 4;               // 0..15: edge slot this iter

    #pragma unroll 4
    for (int it = 0; it < TILE / (BLOCK / LANES_PE); ++it) {
        const int le = it * (BLOCK / LANES_PE) + eLane;   // 0..255 in tile
        const int e  = base + le;
        if (e >= E) break;

        const int   c  = colS[le];          // LDS broadcast across 16 lanes
        const int   r  = rowS[le];
        const float wt = wS[le];

        // 256B coalesced gather of source row (L2-resident)
        const float4 v = ((const float4*)(x + (size_t)c * EMB))[sub];

        // 4 contiguous f32 atomic adds into destination row (L2-side RMW)
        float* dst = xn + (size_t)r * EMB + sub * 4;
        atomicAdd(dst + 0, v.x * wt);
        atomicAdd(dst + 1, v.y * wt);
        atomicAdd(dst + 2, v.z * wt);
        atomicAdd(dst + 3, v.w * wt);
    }
}

// ---------------------------------------------------------------------------
// Host-side orchestration (graph-capture safe: launches only, all on stream)
// ---------------------------------------------------------------------------
extern "C" void kernel_launch(void* const* d_in, const int* in_sizes, int n_in,
                              void* d_out, int out_size, void* d_ws, size_t ws_size,
                              hipStream_t stream) {
    const int*   edge_index = (const int*)d_in[0];    // [2, E] flat
    const float* ew         = (const float*)d_in[1];  // [E]
    const float* ue         = (const float*)d_in[2];  // [N_USERS, 64]
    const float* ie         = (const float*)d_in[3];  // [N_ITEMS, 64]

    const int E          = in_sizes[1];
    const int nUserElems = in_sizes[2];               // N_USERS * 64
    const int nItemElems = in_sizes[3];               // N_ITEMS * 64
    const int totalElems = nUserElems + nItemElems;   // N_TOTAL * 64

    const int* row = edge_index;        // edge_index[0, :]
    const int* col = edge_index + E;    // edge_index[1, :]

    // Workspace: double-buffered layer embeddings (2 x 38.4 MB).
    float* xA  = (float*)d_ws;
    float* xB  = xA + totalElems;
    float* acc = (float*)d_out;         // accumulator lives in the output

    const int nt4 = totalElems / 4;     // EMB=64 -> always divisible by 4
    const int nu4 = nUserElems / 4;

    const dim3 blk(BLOCK);
    const dim3 gridElem((nt4 + BLOCK - 1) / BLOCK);
    const dim3 gridEdge((E + TILE - 1) / TILE);

    // init: xA = acc = concat(ue, ie); xB = 0 (first scatter target)
    lgcn_init_k<<<gridElem, blk, 0, stream>>>(
        (const float4*)ue, (const float4*)ie,
        (float4*)xA, (float4*)acc, (float4*)xB, nu4, nt4);

    float* xin  = xA;
    float* xout = xB;
    for (int layer = 0; layer < 3; ++layer) {
        // xout was zeroed by the previous addzero (or by init for layer 0)
        lgcn_scatter_k<<<gridEdge, blk, 0, stream>>>(row, col, ew, xin, xout, E);

        // acc += xout (scale 0.25 on last layer); zero the now-dead xin,
        // which is the next layer's scatter target. Skip on the last layer.
        const float s   = (layer == 2) ? 0.25f : 1.0f;
        float4*     zb  = (layer == 2) ? nullptr : (float4*)xin;
        lgcn_addzero_k<<<gridElem, blk, 0, stream>>>(
            (float4*)acc, (const float4*)xout, zb, s, nt4);

        float* tmp = xin; xin = xout; xout = tmp;
    }
}